// GNN_node_Virtualnode_45423574122539
// MI455X (gfx1250) — compile-verified
//
#include <hip/hip_runtime.h>
#include <hip/hip_bf16.h>

// ---------------------------------------------------------------------------
// GIN + virtual-node GNN forward for MI455X (gfx1250, wave32).
//
// Matrix work: V_WMMA_F32_16X16X4_F32 (fp32 WMMA). The workload is ~24 GFLOP
// but L2-resident scatter/gather bound, so fp32 WMMA matches reference
// numerics at negligible cost.
//
// v2: block-level tiling to kill redundant L2 traffic:
//   - 8 waves/block, 128 rows/block
//   - A-panel (128 x K) staged in LDS once per block
//   - weight strips (K x 16) staged in LDS once per block, shared by 8 waves
//   - per-wave 16 x NMID intermediate in LDS (D-layout -> A-fragment reshape)
//   => ~208 KB LDS/block (CDNA5 WGP has 320 KB), dynamic LDS allocation.
// ---------------------------------------------------------------------------

typedef float v2f __attribute__((ext_vector_type(2)));
typedef float v8f __attribute__((ext_vector_type(8)));

#define N_NODES 50000
#define N_EDGES 400000
#define N_GRAPHS 64
#define EMB 128

// ---------------------------------------------------------------------------
// Fused  out = (relu?)[bn2?]( relu(bn1(A@W1 + b1)) @ W2 + b2 )
// blockDim = 256 (8 waves); wave w owns rows [blk*128 + w*16, +16).
// Dynamic LDS layout:
//   atile  : 8*16*K floats     (A panel, loaded once per block)
//   bpanel : max(K,NMID)*16    (current weight strip, shared)
//   tbuf   : 8*16*NMID         (per-wave intermediate)
// ---------------------------------------------------------------------------
template <int K, int NMID, int NOUT>
__global__ void __launch_bounds__(256)
mlp2_wmma(const float* __restrict__ A, int M,
          const float* __restrict__ W1, const float* __restrict__ b1,
          const float* __restrict__ g1, const float* __restrict__ bb1,
          const float* __restrict__ W2, const float* __restrict__ b2,
          const float* __restrict__ g2, const float* __restrict__ bb2,
          int relu_out, float* __restrict__ out)
{
    constexpr int WAVES = 8;
    constexpr int ROWS  = WAVES * 16;                       // 128 rows / block
    constexpr int BP    = (K > NMID ? K : NMID) * 16;       // strip floats

    extern __shared__ float smem[];
    float* atile  = smem;                                   // ROWS*K
    float* bpanel = atile + ROWS * K;                       // BP
    float* tbuf   = bpanel + BP;                            // ROWS*NMID

    const int wave = threadIdx.x >> 5;
    const int lane = threadIdx.x & 31;
    const int l16  = lane & 15;
    const int half = lane >> 4;

    const int r0b = blockIdx.x * ROWS;                      // block row base
    const int r0  = r0b + wave * 16;                        // wave row base

    // ---- cooperative load of A panel (clamp OOB rows to a valid row) ----
    for (int i = threadIdx.x * 4; i < ROWS * K; i += 256 * 4) {
        const int r  = i / K;
        const int c  = i - r * K;
        int gr = r0b + r;
        if (gr >= M) gr = M - 1;
        *(float4*)&atile[i] = *(const float4*)&A[(size_t)gr * K + c];
    }

    float* tt = &tbuf[wave * 16 * NMID];
    const float* at = &atile[wave * 16 * K];
    const float bnf = rsqrtf(1.0f + 1e-5f);                 // eval BN: var=1

    // ---- GEMM1: t = relu(bn1(A @ W1 + b1)) -> tbuf ----
    for (int nt = 0; nt < NMID / 16; ++nt) {
        __syncthreads();   // previous strip fully consumed / A panel ready
        // stage W1 strip [K x 16] (columns nt*16..)
        for (int i = threadIdx.x; i < K * 16; i += 256) {
            const int r = i >> 4, c = i & 15;
            bpanel[i] = W1[(size_t)r * NMID + nt * 16 + c];
        }
        if (nt + 1 < NMID / 16)
            __builtin_prefetch(&W1[(size_t)(nt + 1) * 16], 0, 0);  // next strip
        __syncthreads();

        v8f acc = {};
#pragma unroll 4
        for (int k0 = 0; k0 < K; k0 += 4) {
            v2f a, b;
            // A 16x4 fragment: lanes 0-15 = rows, VGPR0/1 = K + 2*half (+1)
            a.x = at[l16 * K + k0 + 2 * half];
            a.y = at[l16 * K + k0 + 2 * half + 1];
            // B 4x16 fragment from LDS strip
            b.x = bpanel[(k0 + 2 * half) * 16 + l16];
            b.y = bpanel[(k0 + 2 * half + 1) * 16 + l16];
            acc = __builtin_amdgcn_wmma_f32_16x16x4_f32(
                false, a, false, b, (short)0, acc, false, false);
        }
        const int col   = nt * 16 + l16;
        const float sc  = g1[col] * bnf;
        const float sh  = bb1[col];
        const float bia = b1[col];
#pragma unroll
        for (int v = 0; v < 8; ++v) {
            const int m = v + 8 * half;            // C/D layout: M = vgpr + 8*half
            float val = (acc[v] + bia) * sc + sh;
            tt[m * NMID + col] = fmaxf(val, 0.0f);
        }
    }

    // ---- GEMM2: out = epilogue(t @ W2 + b2) ----
    for (int nt = 0; nt < NOUT / 16; ++nt) {
        __syncthreads();
        // stage W2 strip [NMID x 16]
        for (int i = threadIdx.x; i < NMID * 16; i += 256) {
            const int r = i >> 4, c = i & 15;
            bpanel[i] = W2[(size_t)r * NOUT + nt * 16 + c];
        }
        if (nt + 1 < NOUT / 16)
            __builtin_prefetch(&W2[(size_t)(nt + 1) * 16], 0, 0);
        __syncthreads();

        v8f acc = {};
#pragma unroll 4
        for (int k0 = 0; k0 < NMID; k0 += 4) {
            v2f a, b;
            a.x = tt[l16 * NMID + k0 + 2 * half];
            a.y = tt[l16 * NMID + k0 + 2 * half + 1];
            b.x = bpanel[(k0 + 2 * half) * 16 + l16];
            b.y = bpanel[(k0 + 2 * half + 1) * 16 + l16];
            acc = __builtin_amdgcn_wmma_f32_16x16x4_f32(
                false, a, false, b, (short)0, acc, false, false);
        }
        const int col   = nt * 16 + l16;
        const float bia = b2[col];
        const float sc  = g2 ? g2[col] * bnf : 1.0f;
        const float sh  = bb2 ? bb2[col] : 0.0f;
#pragma unroll
        for (int v = 0; v < 8; ++v) {
            const int m   = v + 8 * half;
            const int row = r0 + m;
            float val = (acc[v] + bia) * sc + sh;
            if (relu_out) val = fmaxf(val, 0.0f);
            if (row < M) out[(size_t)row * NOUT + col] = val;
        }
    }
}

// ---------------------------------------------------------------------------
// Edge message: msg = relu(h[src] + edge_attr@be_w + be_b); agg[dst] += msg
// One wave per edge; each lane owns 4 contiguous features (512B coalesced).
// ---------------------------------------------------------------------------
__global__ void edge_msg_kernel(const float* __restrict__ h,
                                const int* __restrict__ src,
                                const int* __restrict__ dst,
                                const float* __restrict__ ea,
                                const float* __restrict__ be_w,  // [10 x 128]
                                const float* __restrict__ be_b,  // [128]
                                float* __restrict__ agg)
{
    const int tid = blockIdx.x * blockDim.x + threadIdx.x;
    const int e   = tid >> 5;
    const int c4  = (tid & 31) << 2;
    if (e >= N_EDGES) return;

    const int s = src[e];
    const int d = dst[e];

    float4 emb = *(const float4*)&be_b[c4];
#pragma unroll
    for (int k = 0; k < 10; ++k) {
        const float  av = ea[(size_t)e * 10 + k];     // wave-uniform -> scalarized
        const float4 w  = *(const float4*)&be_w[k * EMB + c4];
        emb.x += av * w.x;  emb.y += av * w.y;
        emb.z += av * w.z;  emb.w += av * w.w;
    }
    const float4 hv = *(const float4*)&h[(size_t)s * EMB + c4];
    float4 m;
    m.x = fmaxf(hv.x + emb.x, 0.0f);
    m.y = fmaxf(hv.y + emb.y, 0.0f);
    m.z = fmaxf(hv.z + emb.z, 0.0f);
    m.w = fmaxf(hv.w + emb.w, 0.0f);

    float* ap = &agg[(size_t)d * EMB + c4];
    atomicAdd(ap + 0, m.x);
    atomicAdd(ap + 1, m.y);
    atomicAdd(ap + 2, m.z);
    atomicAdd(ap + 3, m.w);
}

// h[n] += vn[batch[n]]
__global__ void add_vn_kernel(float* __restrict__ h, const float* __restrict__ vn,
                              const int* __restrict__ batch)
{
    const int tid = blockIdx.x * blockDim.x + threadIdx.x;
    const int n   = tid >> 5;
    const int c4  = (tid & 31) << 2;
    if (n >= N_NODES) return;
    const int g = batch[n];
    float4* hp = (float4*)&h[(size_t)n * EMB + c4];
    const float4 v = *(const float4*)&vn[(size_t)g * EMB + c4];
    float4 t = *hp;
    t.x += v.x; t.y += v.y; t.z += v.z; t.w += v.w;
    *hp = t;
}

// pool[batch[n]] += h[n]   (pool pre-initialized to vn)
__global__ void pool_kernel(const float* __restrict__ h, const int* __restrict__ batch,
                            float* __restrict__ pool)
{
    const int tid = blockIdx.x * blockDim.x + threadIdx.x;
    const int n   = tid >> 5;
    const int c4  = (tid & 31) << 2;
    if (n >= N_NODES) return;
    const int g = batch[n];
    const float4 v = *(const float4*)&h[(size_t)n * EMB + c4];
    float* pp = &pool[(size_t)g * EMB + c4];
    atomicAdd(pp + 0, v.x);
    atomicAdd(pp + 1, v.y);
    atomicAdd(pp + 2, v.z);
    atomicAdd(pp + 3, v.w);
}

// agg[i] = (1+eps)*h[i] + agg[i]   (in place; eps is a device scalar)
__global__ void pre_kernel(const float* __restrict__ h, float* __restrict__ agg,
                           const float* __restrict__ eps_p, size_t n)
{
    const size_t i = (size_t)blockIdx.x * blockDim.x + threadIdx.x;
    if (i >= n) return;
    const float eps = *eps_p;
    agg[i] = (1.0f + eps) * h[i] + agg[i];
}

__global__ void zero_kernel(float* __restrict__ p, size_t n)
{
    const size_t i = (size_t)blockIdx.x * blockDim.x + threadIdx.x;
    if (i < n) p[i] = 0.0f;
}

__global__ void copy_kernel(float* __restrict__ d, const float* __restrict__ s, size_t n)
{
    const size_t i = (size_t)blockIdx.x * blockDim.x + threadIdx.x;
    if (i < n) d[i] = s[i];
}

// vn[g][c] = vn_emb[c]
__global__ void init_vn_kernel(float* __restrict__ vn, const float* __restrict__ emb)
{
    vn[(size_t)blockIdx.x * EMB + threadIdx.x] = emb[threadIdx.x];
}

// ---------------------------------------------------------------------------
// Host launcher
// ---------------------------------------------------------------------------
static inline size_t mlp_shmem_bytes(int K, int NMID)
{
    const int BP = (K > NMID ? K : NMID) * 16;
    return (size_t)(8 * 16 * K + BP + 8 * 16 * NMID) * sizeof(float);
}

extern "C" void kernel_launch(void* const* d_in, const int* in_sizes, int n_in,
                              void* d_out, int out_size, void* d_ws, size_t ws_size,
                              hipStream_t stream)
{
    (void)in_sizes; (void)n_in; (void)out_size; (void)ws_size;

    // ---- inputs (setup_inputs dict order, params flattened in dict order) ----
    const float* x     = (const float*)d_in[0];
    const int*   ei    = (const int*)d_in[1];     // [2, E]
    const float* ea    = (const float*)d_in[2];   // [E, 10]
    const int*   batch = (const int*)d_in[3];     // [N]

    // atom encoder: w1, b1, bn1_g, bn1_b, w2, b2
    const float* a_w1 = (const float*)d_in[4];
    const float* a_b1 = (const float*)d_in[5];
    const float* a_g1 = (const float*)d_in[6];
    const float* a_s1 = (const float*)d_in[7];
    const float* a_w2 = (const float*)d_in[8];
    const float* a_b2 = (const float*)d_in[9];

    // 3 GIN layers x 11: be_w, be_b, eps, mlp_w1, mlp_b1, mlp_bn_g, mlp_bn_b,
    //                    mlp_w2, mlp_b2, bn_g, bn_b
    const int LBASE = 10;
    // 2 VN layers x 8: w1, b1, bn1_g, bn1_b, w2, b2, bn2_g, bn2_b
    const int VBASE = LBASE + 3 * 11;             // 43
    const float* vn_emb = (const float*)d_in[VBASE + 2 * 8];  // d_in[59]

    // ---- workspace ----
    float* ws   = (float*)d_ws;
    float* hA   = ws;                              // 50000*128
    float* hB   = ws + (size_t)N_NODES * EMB;
    float* agg  = ws + (size_t)2 * N_NODES * EMB;
    float* vn   = ws + (size_t)3 * N_NODES * EMB;  // 64*128
    float* pool = vn + (size_t)N_GRAPHS * EMB;

    const size_t nodeElems  = (size_t)N_NODES * EMB;
    const int tiles         = (N_NODES + 15) / 16;           // 3125
    const int mlpBlocks     = (tiles + 7) / 8;               // 8 waves / block
    const int nodeWarpGrid  = (N_NODES * 32 + 255) / 256;
    const int edgeWarpGrid  = (N_EDGES * 32 + 255) / 256;
    const int elemGrid      = (int)((nodeElems + 255) / 256);

    const size_t sh_atom = mlp_shmem_bytes(128, 128);        // ~136 KB
    const size_t sh_gin  = mlp_shmem_bytes(128, 256);        // ~208 KB

    // allow >64KB dynamic LDS (CDNA5 WGP: 320 KB)
    (void)hipFuncSetAttribute((const void*)mlp2_wmma<128, 128, 128>,
                              hipFuncAttributeMaxDynamicSharedMemorySize,
                              (int)sh_atom);
    (void)hipFuncSetAttribute((const void*)mlp2_wmma<128, 256, 128>,
                              hipFuncAttributeMaxDynamicSharedMemorySize,
                              (int)sh_gin);

    // ---- atom encoder: h = relu(bn(x@w1+b1)) @ w2 + b2 ----
    mlp2_wmma<128, 128, 128><<<mlpBlocks, 256, sh_atom, stream>>>(
        x, N_NODES, a_w1, a_b1, a_g1, a_s1, a_w2, a_b2,
        nullptr, nullptr, /*relu_out=*/0, hA);

    // ---- vn = vn_emb broadcast ----
    init_vn_kernel<<<N_GRAPHS, EMB, 0, stream>>>(vn, vn_emb);

    float* hcur  = hA;
    float* hnext = hB;

    for (int l = 0; l < 3; ++l) {
        const float** P = (const float**)&d_in[LBASE + l * 11];
        const float* be_w = P[0];  const float* be_b = P[1];
        const float* epsp = P[2];
        const float* mw1  = P[3];  const float* mb1  = P[4];
        const float* mg1  = P[5];  const float* ms1  = P[6];
        const float* mw2  = P[7];  const float* mb2  = P[8];
        const float* og   = P[9];  const float* os   = P[10];

        // h += vn[batch]
        add_vn_kernel<<<nodeWarpGrid, 256, 0, stream>>>(hcur, vn, batch);
        // agg = 0 ; agg += relu(h[src] + edge_emb) scattered to dst
        zero_kernel<<<elemGrid, 256, 0, stream>>>(agg, nodeElems);
        edge_msg_kernel<<<edgeWarpGrid, 256, 0, stream>>>(
            hcur, ei, ei + N_EDGES, ea, be_w, be_b, agg);
        // agg = (1+eps)*h + agg
        pre_kernel<<<elemGrid, 256, 0, stream>>>(hcur, agg, epsp, nodeElems);

        // h_new = bn(relu(bn(pre@W1+b1)) @ W2 + b2)  [+relu except last layer]
        float* outp = (l == 2) ? (float*)d_out : hnext;
        mlp2_wmma<128, 256, 128><<<mlpBlocks, 256, sh_gin, stream>>>(
            agg, N_NODES, mw1, mb1, mg1, ms1, mw2, mb2, og, os,
            /*relu_out=*/(l < 2) ? 1 : 0, outp);

        if (l < 2) {
            const float** V = (const float**)&d_in[VBASE + l * 8];
            // pool = vn + segment_sum(h, batch)   (h = post-vn-add value)
            copy_kernel<<<(int)((N_GRAPHS * EMB + 255) / 256), 256, 0, stream>>>(
                pool, vn, (size_t)N_GRAPHS * EMB);
            pool_kernel<<<nodeWarpGrid, 256, 0, stream>>>(hcur, batch, pool);
            // vn = relu(bn(relu(bn(pool@Vw1+b1)) @ Vw2 + b2))
            mlp2_wmma<128, 256, 128><<<1, 256, sh_gin, stream>>>(
                pool, N_GRAPHS, V[0], V[1], V[2], V[3], V[4], V[5], V[6], V[7],
                /*relu_out=*/1, vn);

            float* t = hcur; hcur = hnext; hnext = t;
        }
    }
}